// SCSSystem_53781580480530
// MI455X (gfx1250) — compile-verified
//
#include <hip/hip_runtime.h>

// out[b, tidx[n]] += src[b, sidx[n]] * w[n]   (scatter-add, duplicate targets allowed)
// Memory-bound: ~210 MB/call => ~9us floor @ 23.3 TB/s. No contraction dim => no WMMA.
// CDNA5 path: async global->LDS staging of metadata + s_wait_asynccnt, NT spike stream,
// non-returning f32 atomics, ~90% of atomics skipped via spike sparsity.

#define TPB 256

__global__ __launch_bounds__(TPB) void spike_scatter_kernel(
    const float* __restrict__ src,   // [B, SH*SW]
    const float* __restrict__ wts,   // [N]
    const int*   __restrict__ si,    // [N] source indices
    const int*   __restrict__ ti,    // [N] target indices
    const int*   __restrict__ thp,   // scalar TH (device)
    const int*   __restrict__ twp,   // scalar TW (device)
    float* __restrict__ out,         // [B, TH*TW]
    int N, int src_total, int out_total)
{
  __shared__ int   s_si[TPB];
  __shared__ int   s_ti[TPB];
  __shared__ float s_w [TPB];

  const int t = threadIdx.x;
  const long long n  = (long long)blockIdx.x * TPB + t;
  const long long nc = (n < N) ? n : (long long)(N - 1);  // clamp OOB lanes (safe addr)

  // Per-lane LDS byte offsets (low 32 bits of the flat shared address).
  unsigned l_si = (unsigned)(unsigned long long)(void*)&s_si[t];
  unsigned l_ti = (unsigned)(unsigned long long)(void*)&s_ti[t];
  unsigned l_w  = (unsigned)(unsigned long long)(void*)&s_w [t];
  unsigned long long g_si = (unsigned long long)(si  + nc);
  unsigned long long g_ti = (unsigned long long)(ti  + nc);
  unsigned long long g_w  = (unsigned long long)(wts + nc);

  // Async stage metadata into LDS (ASYNCcnt path, off the VGPR-return load pipe).
  asm volatile("global_load_async_to_lds_b32 %0, %1, off" :: "v"(l_si), "v"(g_si) : "memory");
  asm volatile("global_load_async_to_lds_b32 %0, %1, off" :: "v"(l_ti), "v"(g_ti) : "memory");
  asm volatile("global_load_async_to_lds_b32 %0, %1, off" :: "v"(l_w ), "v"(g_w ) : "memory");

  // Uniform shape math while the async copies are in flight.
  const int TH = *thp;
  const int TW = *twp;
  const int tplane = TH * TW;
  const int B      = (tplane > 0) ? (out_total / tplane) : 0;
  const int splane = (B > 0) ? (src_total / B) : 0;

  // Each lane only reads its own slot, written by its own wave's async ops.
  asm volatile("s_wait_asynccnt 0" ::: "memory");

  const int   sidx = s_si[t];
  const int   tidx = s_ti[t];
  const float wt   = s_w [t];

  if (n < N) {
    const float* sp = src + sidx;
    float*       op = out + tidx;
#pragma unroll 4
    for (int b = 0; b < B; ++b) {
      // Spike stream: read-once -> non-temporal, keep L2 for the output atomics.
      float v = __builtin_nontemporal_load(sp + (long long)b * splane);
      if (v != 0.0f) {                         // ~90% of lanes skip (SPIKE_PROB=0.1)
        atomicAdd(op + (long long)b * tplane, v * wt);  // non-returning f32 atomic
      }
    }
  }
}

// Vectorized zero-fill for the bulk (n4 float4 stores) ...
__global__ void zero4_kernel(float4* __restrict__ out4, int n4) {
  int i = blockIdx.x * blockDim.x + threadIdx.x;
  if (i < n4) out4[i] = make_float4(0.f, 0.f, 0.f, 0.f);
}

// ... and a tiny scalar tail (0-3 elements).
__global__ void zero_tail_kernel(float* __restrict__ out, long long start, long long n) {
  long long i = start + threadIdx.x;
  if (i < n) out[i] = 0.f;
}

extern "C" void kernel_launch(void* const* d_in, const int* in_sizes, int n_in,
                              void* d_out, int out_size, void* d_ws, size_t ws_size,
                              hipStream_t stream) {
  const float* src = (const float*)d_in[0];   // [B*SH*SW] f32
  const float* wts = (const float*)d_in[1];   // [N] f32
  const int*   si  = (const int*)  d_in[2];   // [N] i32
  const int*   ti  = (const int*)  d_in[3];   // [N] i32
  const int*   thp = (const int*)  d_in[4];   // scalar
  const int*   twp = (const int*)  d_in[5];   // scalar
  float*       out = (float*)d_out;

  const int N         = in_sizes[1];
  const int src_total = in_sizes[0];
  const int out_total = out_size;

  // 1) zero-init output (harness poisons it)
  int n4 = out_total / 4;
  if (n4 > 0) {
    int zblocks = (n4 + TPB - 1) / TPB;
    zero4_kernel<<<zblocks, TPB, 0, stream>>>((float4*)out, n4);
  }
  int tail = out_total - n4 * 4;
  if (tail > 0) {
    zero_tail_kernel<<<1, 4, 0, stream>>>(out, (long long)n4 * 4, (long long)out_total);
  }

  // 2) gather * weight -> scatter-add, one thread per connection, B-loop inside
  int sblocks = (N + TPB - 1) / TPB;
  spike_scatter_kernel<<<sblocks, TPB, 0, stream>>>(src, wts, si, ti, thp, twp,
                                                    out, N, src_total, out_total);
}